// Decoder_9715216023709
// MI455X (gfx1250) — compile-verified
//
#include <hip/hip_runtime.h>
#include <hip/hip_bf16.h>

#define N_NODES   20000
#define BATCH     512
#define EMBED_DIM 32
#define NUM_SAMPLES 5001
#define INV_TD_MAX (1.0f/5000.0f)

typedef __attribute__((ext_vector_type(2))) float v2f;
typedef __attribute__((ext_vector_type(8))) float v8f;

__device__ __forceinline__ float softplus_f(float x) {
    // numerically safe softplus: max(x,0) + log(1+exp(-|x|))
    return fmaxf(x, 0.0f) + __logf(1.0f + __expf(-fabsf(x)));
}

// ---------------------------------------------------------------------------
// Kernel 1: node projections via V_WMMA_F32_16X16X4_F32.
//   P(16x16) = E_tile(16x32) @ Wpad(32x16), K-loop of 8 (K=4 per step).
//   Wpad columns: 0=W0u, 1=W1u, 2=W0v, 3=W1v, 4..15 = 0.
// A-matrix f32 16x4 layout: lane l (m=l&15, kh=l>>4): VGPR0=K(2kh), VGPR1=K(2kh+1)
// B-matrix f32 4x16 layout: lane l (n=l&15, kh=l>>4): VGPR0=row(2kh), VGPR1=row(2kh+1)
// D/C f32 16x16 layout: VGPR j, lane l: M = j + 8*(l>>4), N = l&15
//
// B operands are built branch-free (pointer/offset select + value cndmask)
// so the unrolled WMMA loop has no EXEC-mask churn.
// ---------------------------------------------------------------------------
__global__ void __launch_bounds__(32)
project_kernel(const float* __restrict__ E,
               const float* __restrict__ W0,   // 64 floats: [W0u | W0v]
               const float* __restrict__ W1,   // 64 floats: [W1u | W1v]
               float* __restrict__ nodeOut)    // 4 arrays of N_NODES: U0,U1,V0,V1
{
    const int tile = blockIdx.x;            // 20000/16 = 1250 tiles
    const int lane = threadIdx.x;           // 0..31
    const int m  = lane & 15;
    const int kh = lane >> 4;               // 0 or 1
    const int n  = m;                       // B column for this lane

    const float* row = E + (size_t)(tile * 16 + m) * EMBED_DIM;

    // Branch-free B source: column n<4 -> {W0u,W1u,W0v,W1v}, else zeroed.
    const float* Wsel = (n & 1) ? W1 : W0;
    const int    woff = (n & 2) ? EMBED_DIM : 0;   // u-half vs v-half
    const float  msk  = (n < 4) ? 1.0f : 0.0f;     // value select, no EXEC change

    v8f c = {};
#pragma unroll
    for (int kk = 0; kk < 8; ++kk) {
        const int k0 = kk * 4 + 2 * kh;            // even -> 8B-aligned v2f loads
        const v2f a  = *(const v2f*)(row + k0);
        v2f bm       = *(const v2f*)(Wsel + woff + k0);
        bm.x *= msk;
        bm.y *= msk;

        c = __builtin_amdgcn_wmma_f32_16x16x4_f32(
                /*neg_a=*/false, a, /*neg_b=*/false, bm,
                /*c_mod=*/(short)0, c, /*reuse_a=*/false, /*reuse_b=*/false);
    }

    // Columns 0..3 hold the four projections for the 16 rows of this tile.
    if (n < 4) {
        const int rowBase = tile * 16 + 8 * kh;    // VGPR j holds M = j + 8*kh
        float* dst = nodeOut + (size_t)n * N_NODES;
#pragma unroll
        for (int j = 0; j < 8; ++j) dst[rowBase + j] = c[j];
    }
}

// ---------------------------------------------------------------------------
// Kernel 2: per-batch scalars (fold Hawkes additive term into a constant)
// ---------------------------------------------------------------------------
__global__ void __launch_bounds__(256)
batch_scalars(const float* __restrict__ E, const int* __restrict__ assoc,
              const int* __restrict__ src, const int* __restrict__ pos_dst,
              const float* __restrict__ last_update, const float* __restrict__ cur_time,
              const int* __restrict__ et,
              const float* __restrict__ W0, const float* __restrict__ b0,
              const float* __restrict__ W1, const float* __restrict__ b1,
              const float* __restrict__ psi, const float* __restrict__ alpha,
              const float* __restrict__ w_t,
              float* __restrict__ sS, float* __restrict__ sD,
              float* __restrict__ psiB, float* __restrict__ invPsiB,
              float* __restrict__ gbB, float* __restrict__ alphaB,
              float* __restrict__ wtB, int* __restrict__ etbB)
{
    const int b = blockIdx.x * blockDim.x + threadIdx.x;
    if (b >= BATCH) return;

    const int e = (et[b] > 0) ? 1 : 0;
    const float* W  = e ? W1 : W0;
    const float  bb = e ? b1[0] : b0[0];

    const int is = assoc[src[b]];
    const int id = assoc[pos_dst[b]];
    const float* zs = E + (size_t)is * EMBED_DIM;
    const float* zd = E + (size_t)id * EMBED_DIM;

    float du = 0.0f, dv = 0.0f;
#pragma unroll 8
    for (int k = 0; k < EMBED_DIM; ++k) {
        du += zs[k] * W[k];                 // z_src . Wu_sel
        dv += zd[k] * W[EMBED_DIM + k];     // z_dst . Wv_sel
    }

    const float ps = psi[e], al = alpha[e], wt = w_t[e];
    const float ct = cur_time[b];
    const float tds = (ct - last_update[is]) * INV_TD_MAX;
    const float tdd = (ct - last_update[id]) * INV_TD_MAX;

    sS[b]      = du + bb + al * __expf(-wt * tds);
    sD[b]      = dv + bb + al * __expf(-wt * tdd);
    psiB[b]    = ps;
    invPsiB[b] = 1.0f / (ps + 1e-7f);
    gbB[b]     = du + dv + bb;
    alphaB[b]  = al;
    wtB[b]     = wt;
    etbB[b]    = e;
}

// ---------------------------------------------------------------------------
// Kernel 3: lambda_src / lambda_dst (store-bound, float4 path)
// ---------------------------------------------------------------------------
__global__ void __launch_bounds__(256)
lambda_kernel(const float* __restrict__ nodeU0, const float* __restrict__ nodeU1,
              const float* __restrict__ nodeV0, const float* __restrict__ nodeV1,
              const float* __restrict__ sS, const float* __restrict__ sD,
              const float* __restrict__ psiB, const float* __restrict__ invPsiB,
              const int* __restrict__ etbB,
              float* __restrict__ out)
{
    const int b = blockIdx.y;
    const int n = (blockIdx.x * blockDim.x + threadIdx.x) * 4;
    if (n >= N_NODES) return;   // N_NODES % 4 == 0 -> full float4 or fully out

    const int e = etbB[b];
    const float4* v4 = (const float4*)(e ? nodeV1 : nodeV0) + (n >> 2);
    const float4* u4 = (const float4*)(e ? nodeU1 : nodeU0) + (n >> 2);
    const float s1 = sS[b], s2 = sD[b], ps = psiB[b], ip = invPsiB[b];

    const float4 vv = *v4;
    const float4 uu = *u4;

    float4 ls, ld;
    ls.x = ps * softplus_f((s1 + vv.x) * ip);
    ls.y = ps * softplus_f((s1 + vv.y) * ip);
    ls.z = ps * softplus_f((s1 + vv.z) * ip);
    ls.w = ps * softplus_f((s1 + vv.w) * ip);
    ld.x = ps * softplus_f((s2 + uu.x) * ip);
    ld.y = ps * softplus_f((s2 + uu.y) * ip);
    ld.z = ps * softplus_f((s2 + uu.z) * ip);
    ld.w = ps * softplus_f((s2 + uu.w) * ip);

    const size_t BN = (size_t)BATCH * N_NODES;
    *(float4*)(out + (size_t)b * N_NODES + n)      = ls;
    *(float4*)(out + BN + (size_t)b * N_NODES + n) = ld;
}

// ---------------------------------------------------------------------------
// Kernel 4: return_time_pred — one wave per batch row, shuffle-scan cumsum
// ---------------------------------------------------------------------------
__global__ void __launch_bounds__(256)
rtp_kernel(const float* __restrict__ gbB, const float* __restrict__ psiB,
           const float* __restrict__ invPsiB, const float* __restrict__ alphaB,
           const float* __restrict__ wtB,
           float* __restrict__ out)   // BATCH elements
{
    const int wave = (blockIdx.x * blockDim.x + threadIdx.x) >> 5;
    const int lane = threadIdx.x & 31;
    if (wave >= BATCH) return;

    const float gb = gbB[wave], ps = psiB[wave], ip = invPsiB[wave];
    const float al = alphaB[wave], wt = wtB[wave];

    float carry = 0.0f;   // running integral up to previous chunk
    float acc   = 0.0f;   // weighted trapezoid accumulator

    for (int base = 0; base < NUM_SAMPLES; base += 32) {
        const int i = base + lane;
        float lam = 0.0f;
        if (i < NUM_SAMPLES) {
            const float td = (float)i * INV_TD_MAX;       // lin[i] / TRAIN_TD_MAX
            const float g  = gb + al * __expf(-wt * td);
            lam = ps * softplus_f(g * ip);
        }
        // inclusive prefix sum of lam over the wave
        float x = lam;
#pragma unroll
        for (int off = 1; off < 32; off <<= 1) {
            float y = __shfl_up(x, off, 32);
            if (lane >= off) x += y;
        }
        const float integral = carry + x;                 // cumsum(TIMESTEP * intensity)
        const float dens = lam * __expf(-integral);
        const float t    = (float)i * dens;               // lin[i] * density
        // trapezoid weights: t_0 and t_{S-1} get 0.5, interior 1.0 (t_0==0 anyway)
        float w = 1.0f;
        if (i >= NUM_SAMPLES - 1) w = (i == NUM_SAMPLES - 1) ? 0.5f : 0.0f;
        acc += w * t;
        carry += __shfl(x, 31, 32);                       // total of this chunk
    }

    // wave reduction of acc
#pragma unroll
    for (int off = 16; off >= 1; off >>= 1) acc += __shfl_xor(acc, off, 32);
    if (lane == 0) out[wave] = acc;                       // TIMESTEP == 1
}

// ---------------------------------------------------------------------------
extern "C" void kernel_launch(void* const* d_in, const int* in_sizes, int n_in,
                              void* d_out, int out_size, void* d_ws, size_t ws_size,
                              hipStream_t stream) {
    const float* E           = (const float*)d_in[0];
    const int*   assoc       = (const int*)  d_in[1];
    const int*   src         = (const int*)  d_in[2];
    const int*   pos_dst     = (const int*)  d_in[3];
    // d_in[4] = neg_dst (unused by reference outputs)
    const float* last_update = (const float*)d_in[5];
    const float* cur_time    = (const float*)d_in[6];
    const int*   et          = (const int*)  d_in[7];
    const float* W0          = (const float*)d_in[8];
    const float* b0          = (const float*)d_in[9];
    const float* W1          = (const float*)d_in[10];
    const float* b1          = (const float*)d_in[11];
    const float* psi         = (const float*)d_in[12];
    const float* alpha       = (const float*)d_in[13];
    const float* w_t         = (const float*)d_in[14];

    float* out = (float*)d_out;

    // workspace layout
    float* ws      = (float*)d_ws;
    float* nodeArr = ws;                       // 4 * N_NODES (U0,U1,V0,V1)
    float* pb      = ws + 4 * N_NODES;
    float* sS      = pb + 0 * BATCH;
    float* sD      = pb + 1 * BATCH;
    float* psiB    = pb + 2 * BATCH;
    float* invPsiB = pb + 3 * BATCH;
    float* gbB     = pb + 4 * BATCH;
    float* alphaB  = pb + 5 * BATCH;
    float* wtB     = pb + 6 * BATCH;
    int*   etbB    = (int*)(pb + 7 * BATCH);

    // 1) WMMA node projections: 1250 tiles, one wave each
    project_kernel<<<N_NODES / 16, 32, 0, stream>>>(E, W0, W1, nodeArr);

    // 2) per-batch scalars
    batch_scalars<<<(BATCH + 255) / 256, 256, 0, stream>>>(
        E, assoc, src, pos_dst, last_update, cur_time, et,
        W0, b0, W1, b1, psi, alpha, w_t,
        sS, sD, psiB, invPsiB, gbB, alphaB, wtB, etbB);

    // 3) lambda_src / lambda_dst (B x N each)
    dim3 grid((N_NODES / 4 + 255) / 256, BATCH);
    lambda_kernel<<<grid, 256, 0, stream>>>(
        nodeArr + 0 * N_NODES, nodeArr + 1 * N_NODES,
        nodeArr + 2 * N_NODES, nodeArr + 3 * N_NODES,
        sS, sD, psiB, invPsiB, etbB, out);

    // 4) return_time_pred (one wave per batch row)
    const size_t BN = (size_t)BATCH * N_NODES;
    rtp_kernel<<<(BATCH * 32) / 256, 256, 0, stream>>>(
        gbB, psiB, invPsiB, alphaB, wtB, out + 2 * BN);
}